// SOMM_v2_29429115912492
// MI455X (gfx1250) — compile-verified
//
#include <hip/hip_runtime.h>
#include <math.h>

// ---- problem constants (from setup_inputs) ----
#define NB 64      // batch
#define NC 1024    // channels
#define ND 128     // att_dim
#define NR 256     // HW = sp_reso
#define BN_EPS 1e-5f

typedef __attribute__((ext_vector_type(2))) float v2f;
typedef __attribute__((ext_vector_type(8))) float v8f;

union V8 { v8f v; float f[8]; };

__device__ __forceinline__ v8f wmma4(v2f a, v2f b, v8f c) {
  // D = A(16x4) * B(4x16) + C(16x16), full fp32 (V_WMMA_F32_16X16X4_F32)
  return __builtin_amdgcn_wmma_f32_16x16x4_f32(
      /*neg_a=*/false, a, /*neg_b=*/false, b,
      /*c_mod=*/(short)0, c, /*reuse_a=*/false, /*reuse_b=*/false);
}

// ---------------------------------------------------------------------------
// Precompute fused BN1 (with conv bias folded) and BN2+rowconv coefficients.
// ---------------------------------------------------------------------------
__global__ __launch_bounds__(256) void prep_kernel(
    const float* __restrict__ b_dr,
    const float* __restrict__ g1, const float* __restrict__ be1,
    const float* __restrict__ m1, const float* __restrict__ v1,
    const float* __restrict__ g2, const float* __restrict__ be2,
    const float* __restrict__ m2, const float* __restrict__ v2,
    const float* __restrict__ rcw, const float* __restrict__ rcb,
    float* __restrict__ sc1, float* __restrict__ sh1,
    float* __restrict__ rsc, float* __restrict__ rsh)
{
  int i = threadIdx.x;
  if (i < ND) {
    float s = g1[i] * rsqrtf(v1[i] + BN_EPS);
    sc1[i] = s;
    sh1[i] = (b_dr[i] - m1[i]) * s + be1[i];
  }
  float s2 = g2[i] * rsqrtf(v2[i] + BN_EPS);
  rsc[i] = s2 * rcw[i];
  rsh[i] = (be2[i] - m2[i] * s2) * rcw[i] + rcb[i];
}

// ---------------------------------------------------------------------------
// GEMM1: y(128x256) = w_dr(128x1024) @ x_b(1024x256); epilogue BN1+ReLU,
// stored TRANSPOSED as z[b][p][d] (p=HW position, d=att dim).
// grid = NB * 8 (2x4 tiles of 64x64 per batch), block = 256.
// ---------------------------------------------------------------------------
__global__ __launch_bounds__(256) void gemm1_kernel(
    const float* __restrict__ x, const float* __restrict__ w,
    const float* __restrict__ sc1, const float* __restrict__ sh1,
    float* __restrict__ z)
{
  const int tid = threadIdx.x;
  const int lane = tid & 31, wv = tid >> 5;
  const int batch = blockIdx.x >> 3;
  const int t = blockIdx.x & 7;
  const int m0 = (t >> 2) * 64;   // D rows
  const int n0 = (t & 3) * 64;    // P cols
  const float* Bx = x + (size_t)batch * (NC * NR);

  __shared__ float As[64 * 17];
  __shared__ float Bs[16 * 65];

  const int t0 = wv, t1 = wv + 8;
  const int tm0 = t0 >> 2, tn0 = t0 & 3, tm1 = t1 >> 2, tn1 = t1 & 3;
  const int lr = lane & 15;
  const int ah = (lane >> 4) * 2;  // A k sub-offset
  const int bh = (lane >> 4);      // B k sub-offset
  v8f acc0 = {}; v8f acc1 = {};

  for (int kk = 0; kk < NC; kk += 16) {
    __syncthreads();
    {
      int r = tid >> 2, kq = (tid & 3) * 4;
      float4 va = *(const float4*)(w + (size_t)(m0 + r) * NC + kk + kq);
      As[r * 17 + kq + 0] = va.x; As[r * 17 + kq + 1] = va.y;
      As[r * 17 + kq + 2] = va.z; As[r * 17 + kq + 3] = va.w;
      int kr = tid >> 4, cq = (tid & 15) * 4;
      float4 vb = *(const float4*)(Bx + (size_t)(kk + kr) * NR + n0 + cq);
      Bs[kr * 65 + cq + 0] = vb.x; Bs[kr * 65 + cq + 1] = vb.y;
      Bs[kr * 65 + cq + 2] = vb.z; Bs[kr * 65 + cq + 3] = vb.w;
    }
    __syncthreads();
#pragma unroll
    for (int ks = 0; ks < 16; ks += 4) {
      v2f a0, a1, b0, b1;
      a0.x = As[(tm0 * 16 + lr) * 17 + ks + ah];
      a0.y = As[(tm0 * 16 + lr) * 17 + ks + ah + 1];
      b0.x = Bs[(ks + bh) * 65 + tn0 * 16 + lr];
      b0.y = Bs[(ks + bh + 2) * 65 + tn0 * 16 + lr];
      acc0 = wmma4(a0, b0, acc0);
      a1.x = As[(tm1 * 16 + lr) * 17 + ks + ah];
      a1.y = As[(tm1 * 16 + lr) * 17 + ks + ah + 1];
      b1.x = Bs[(ks + bh) * 65 + tn1 * 16 + lr];
      b1.y = Bs[(ks + bh + 2) * 65 + tn1 * 16 + lr];
      acc1 = wmma4(a1, b1, acc1);
    }
  }
  float* zb = z + (size_t)batch * (NR * ND);
  const int rhalf = (lane >> 4) * 8;
  V8 u0; u0.v = acc0; V8 u1; u1.v = acc1;
#pragma unroll
  for (int r = 0; r < 8; ++r) {
    int d = m0 + tm0 * 16 + r + rhalf, p = n0 + tn0 * 16 + lr;
    zb[p * ND + d] = fmaxf(fmaf(u0.f[r], sc1[d], sh1[d]), 0.f);
    d = m0 + tm1 * 16 + r + rhalf; p = n0 + tn1 * 16 + lr;
    zb[p * ND + d] = fmaxf(fmaf(u1.f[r], sc1[d], sh1[d]), 0.f);
  }
}

// ---------------------------------------------------------------------------
// Row means of z over the D=128 axis: mu[b*256+p]. One wave per row.
// ---------------------------------------------------------------------------
__global__ __launch_bounds__(256) void mean_kernel(
    const float* __restrict__ z, float* __restrict__ mu)
{
  const int lane = threadIdx.x & 31, wv = threadIdx.x >> 5;
  int row = blockIdx.x * 8 + wv;  // NB*NR = 16384 rows
  float4 v = *(const float4*)(z + (size_t)row * ND + lane * 4);
  float s = v.x + v.y + v.z + v.w;
#pragma unroll
  for (int o = 16; o > 0; o >>= 1) s += __shfl_down(s, o);
  if (lane == 0) mu[row] = s * (1.0f / ND);
}

// ---------------------------------------------------------------------------
// cov[b] = (1/128) z_b z_b^T - mu mu^T   (256x256). grid = NB*16 blocks.
// ---------------------------------------------------------------------------
__global__ __launch_bounds__(256) void cov_kernel(
    const float* __restrict__ z, const float* __restrict__ mu,
    float* __restrict__ cov)
{
  const int tid = threadIdx.x;
  const int lane = tid & 31, wv = tid >> 5;
  const int batch = blockIdx.x >> 4;
  const int t = blockIdx.x & 15;
  const int m0 = (t >> 2) * 64, n0 = (t & 3) * 64;
  const float* zb = z + (size_t)batch * (NR * ND);

  __shared__ float As[64 * 17];
  __shared__ float Bs[16 * 65];

  const int t0 = wv, t1 = wv + 8;
  const int tm0 = t0 >> 2, tn0 = t0 & 3, tm1 = t1 >> 2, tn1 = t1 & 3;
  const int lr = lane & 15;
  const int ah = (lane >> 4) * 2, bh = (lane >> 4);
  v8f acc0 = {}; v8f acc1 = {};

  for (int kk = 0; kk < ND; kk += 16) {
    __syncthreads();
    {
      int r = tid >> 2, kq = (tid & 3) * 4;
      float4 va = *(const float4*)(zb + (size_t)(m0 + r) * ND + kk + kq);
      As[r * 17 + kq + 0] = va.x; As[r * 17 + kq + 1] = va.y;
      As[r * 17 + kq + 2] = va.z; As[r * 17 + kq + 3] = va.w;
      // B = z^T: Bs[k][n] = z[n0+n][k]
      float4 vb = *(const float4*)(zb + (size_t)(n0 + r) * ND + kk + kq);
      Bs[(kq + 0) * 65 + r] = vb.x; Bs[(kq + 1) * 65 + r] = vb.y;
      Bs[(kq + 2) * 65 + r] = vb.z; Bs[(kq + 3) * 65 + r] = vb.w;
    }
    __syncthreads();
#pragma unroll
    for (int ks = 0; ks < 16; ks += 4) {
      v2f a0, a1, b0, b1;
      a0.x = As[(tm0 * 16 + lr) * 17 + ks + ah];
      a0.y = As[(tm0 * 16 + lr) * 17 + ks + ah + 1];
      b0.x = Bs[(ks + bh) * 65 + tn0 * 16 + lr];
      b0.y = Bs[(ks + bh + 2) * 65 + tn0 * 16 + lr];
      acc0 = wmma4(a0, b0, acc0);
      a1.x = As[(tm1 * 16 + lr) * 17 + ks + ah];
      a1.y = As[(tm1 * 16 + lr) * 17 + ks + ah + 1];
      b1.x = Bs[(ks + bh) * 65 + tn1 * 16 + lr];
      b1.y = Bs[(ks + bh + 2) * 65 + tn1 * 16 + lr];
      acc1 = wmma4(a1, b1, acc1);
    }
  }
  float* cb = cov + (size_t)batch * (NR * NR);
  const float* mb = mu + batch * NR;
  const int rhalf = (lane >> 4) * 8;
  V8 u0; u0.v = acc0; V8 u1; u1.v = acc1;
#pragma unroll
  for (int r = 0; r < 8; ++r) {
    int gr = m0 + tm0 * 16 + r + rhalf, gc = n0 + tn0 * 16 + lr;
    cb[gr * NR + gc] = u0.f[r] * (1.0f / ND) - mb[gr] * mb[gc];
    gr = m0 + tm1 * 16 + r + rhalf; gc = n0 + tn1 * 16 + lr;
    cb[gr * NR + gc] = u1.f[r] * (1.0f / ND) - mb[gr] * mb[gc];
  }
}

// ---------------------------------------------------------------------------
// normA = trace(cov)/3 per batch; store 1/normA and sqrt(normA).
// ---------------------------------------------------------------------------
__global__ __launch_bounds__(256) void trace_kernel(
    const float* __restrict__ cov, float* __restrict__ inorm,
    float* __restrict__ snorm)
{
  __shared__ float red[256];
  int b = blockIdx.x;
  red[threadIdx.x] = cov[(size_t)b * (NR * NR) + threadIdx.x * (NR + 1)];
  __syncthreads();
  for (int s = 128; s > 0; s >>= 1) {
    if (threadIdx.x < s) red[threadIdx.x] += red[threadIdx.x + s];
    __syncthreads();
  }
  if (threadIdx.x == 0) {
    float normA = fmaxf(red[0] * (1.0f / 3.0f), 1e-30f);
    inorm[b] = 1.0f / normA;
    snorm[b] = sqrtf(normA);
  }
}

// ---------------------------------------------------------------------------
// Z0 = 0.5*(3I - A/normA)   (elementwise + diagonal). float4 per thread.
// ---------------------------------------------------------------------------
__global__ __launch_bounds__(256) void initz_kernel(
    const float* __restrict__ A, const float* __restrict__ inorm,
    float* __restrict__ Z)
{
  int idx = blockIdx.x * 256 + threadIdx.x;  // NB*NR*NR/4 = 1048576
  int base = idx * 4;
  int b = base >> 16;
  int ij = base & 65535;
  int i = ij >> 8, j = ij & 255;
  float s = -0.5f * inorm[b];
  float4 a = *(const float4*)(A + base);
  float4 o;
  o.x = s * a.x + ((i == j + 0) ? 1.5f : 0.f);
  o.y = s * a.y + ((i == j + 1) ? 1.5f : 0.f);
  o.z = s * a.z + ((i == j + 2) ? 1.5f : 0.f);
  o.w = s * a.w + ((i == j + 3) ? 1.5f : 0.f);
  *(float4*)(Z + base) = o;
}

// ---------------------------------------------------------------------------
// Batched 256x256x256 matmul: C = mulc * s_b * (A@B) + diagc * I.
// Dominant Newton-Schulz kernel. 128x128 block tile, 256 threads (8 waves),
// each wave owns a 64x32 C-region (4x2 tiles of 16x16 -> 8 accumulators).
// Per k-step: 4 A-frags + 2 B-frags feed 8 WMMAs (2x the wmma:ds ratio of a
// 64x64 tiling). LDS: A 128x16 pitch 20 (10 KB), B 16x128 pitch 132 (8.25 KB).
//   T = 0.5(3I - Z@Y)  ->  mulc=-0.5, diagc=1.5
//   Y' = Y@T, Z' = T@Z ->  mulc=1,    diagc=0
// grid = NB*4.
// ---------------------------------------------------------------------------
#define AP 20    // As pitch (gcd(20,64)=4 -> conflict-free row reads)
#define BP 132   // Bs pitch

__device__ __forceinline__ void bmm128_core(
    const float* __restrict__ Ab, const float* __restrict__ Bb,
    float* As, float* Bs, v8f acc[4][2],
    int tid, int lane, int m0, int n0)
{
  const int wv = tid >> 5;
  const int lr = lane & 15;
  const int ah = (lane >> 4) * 2, bh = (lane >> 4);
  const int wrow = (wv >> 2) * 64;   // 0 or 64
  const int wcol = (wv & 3) * 32;    // 0,32,64,96
  const int ar = tid >> 1, akq = (tid & 1) * 8;      // A panel: 2 float4/thread
  const int bkr = tid >> 4, bcq = (tid & 15) * 8;    // B panel: 2 float4/thread

  for (int kk = 0; kk < NR; kk += 16) {
    __syncthreads();
    {
      float4 va0 = *(const float4*)(Ab + (size_t)(m0 + ar) * NR + kk + akq);
      float4 va1 = *(const float4*)(Ab + (size_t)(m0 + ar) * NR + kk + akq + 4);
      *(float4*)&As[ar * AP + akq] = va0;
      *(float4*)&As[ar * AP + akq + 4] = va1;
      float4 vb0 = *(const float4*)(Bb + (size_t)(kk + bkr) * NR + n0 + bcq);
      float4 vb1 = *(const float4*)(Bb + (size_t)(kk + bkr) * NR + n0 + bcq + 4);
      *(float4*)&Bs[bkr * BP + bcq] = vb0;
      *(float4*)&Bs[bkr * BP + bcq + 4] = vb1;
    }
    __syncthreads();
#pragma unroll
    for (int ks = 0; ks < 16; ks += 4) {
      v2f af[4], bf[2];
#pragma unroll
      for (int tr = 0; tr < 4; ++tr) {
        af[tr].x = As[(wrow + tr * 16 + lr) * AP + ks + ah];
        af[tr].y = As[(wrow + tr * 16 + lr) * AP + ks + ah + 1];
      }
#pragma unroll
      for (int tc = 0; tc < 2; ++tc) {
        bf[tc].x = Bs[(ks + bh) * BP + wcol + tc * 16 + lr];
        bf[tc].y = Bs[(ks + bh + 2) * BP + wcol + tc * 16 + lr];
      }
#pragma unroll
      for (int tr = 0; tr < 4; ++tr)
#pragma unroll
        for (int tc = 0; tc < 2; ++tc)
          acc[tr][tc] = wmma4(af[tr], bf[tc], acc[tr][tc]);
    }
  }
}

__global__ __launch_bounds__(256) void bmm256_kernel(
    const float* __restrict__ A, const float* __restrict__ Bm,
    float* __restrict__ C, const float* __restrict__ sbuf,
    float mulc, float diagc)
{
  const int tid = threadIdx.x;
  const int lane = tid & 31, wv = tid >> 5;
  const int batch = blockIdx.x >> 2;
  const int t = blockIdx.x & 3;
  const int m0 = (t >> 1) * 128, n0 = (t & 1) * 128;
  const float* Ab = A + (size_t)batch * (NR * NR);
  const float* Bb = Bm + (size_t)batch * (NR * NR);

  __shared__ float As[128 * AP];
  __shared__ float Bs[16 * BP];

  v8f acc[4][2] = {};
  bmm128_core(Ab, Bb, As, Bs, acc, tid, lane, m0, n0);

  float* Cb = C + (size_t)batch * (NR * NR);
  float sm = mulc * (sbuf ? sbuf[batch] : 1.0f);
  const int lr = lane & 15;
  const int rhalf = (lane >> 4) * 8;
  const int wrow = (wv >> 2) * 64, wcol = (wv & 3) * 32;
#pragma unroll
  for (int tr = 0; tr < 4; ++tr) {
#pragma unroll
    for (int tc = 0; tc < 2; ++tc) {
      V8 u; u.v = acc[tr][tc];
#pragma unroll
      for (int r = 0; r < 8; ++r) {
        int gr = m0 + wrow + tr * 16 + r + rhalf;
        int gc = n0 + wcol + tc * 16 + lr;
        Cb[gr * NR + gc] = u.f[r] * sm + ((gr == gc) ? diagc : 0.f);
      }
    }
  }
}

// ---------------------------------------------------------------------------
// Final NS step S = (Y@T)*sqrt(normA), fused with BN2 + per-row conv:
//   out[p][q] = val*rsc[p] + rsh[p].  Same 128x128 core. grid = NB*4.
// ---------------------------------------------------------------------------
__global__ __launch_bounds__(256) void bmm_final_kernel(
    const float* __restrict__ A, const float* __restrict__ Bm,
    float* __restrict__ C, const float* __restrict__ snorm,
    const float* __restrict__ rsc, const float* __restrict__ rsh)
{
  const int tid = threadIdx.x;
  const int lane = tid & 31, wv = tid >> 5;
  const int batch = blockIdx.x >> 2;
  const int t = blockIdx.x & 3;
  const int m0 = (t >> 1) * 128, n0 = (t & 1) * 128;
  const float* Ab = A + (size_t)batch * (NR * NR);
  const float* Bb = Bm + (size_t)batch * (NR * NR);

  __shared__ float As[128 * AP];
  __shared__ float Bs[16 * BP];

  v8f acc[4][2] = {};
  bmm128_core(Ab, Bb, As, Bs, acc, tid, lane, m0, n0);

  float* Cb = C + (size_t)batch * (NR * NR);
  float sn = snorm[batch];
  const int lr = lane & 15;
  const int rhalf = (lane >> 4) * 8;
  const int wrow = (wv >> 2) * 64, wcol = (wv & 3) * 32;
#pragma unroll
  for (int tr = 0; tr < 4; ++tr) {
#pragma unroll
    for (int tc = 0; tc < 2; ++tc) {
      V8 u; u.v = acc[tr][tc];
#pragma unroll
      for (int r = 0; r < 8; ++r) {
        int gr = m0 + wrow + tr * 16 + r + rhalf;
        int gc = n0 + wcol + tc * 16 + lr;
        Cb[gr * NR + gc] = fmaf(u.f[r] * sn, rsc[gr], rsh[gr]);
      }
    }
  }
}

// ---------------------------------------------------------------------------
// FC layer: acc(64x256) += S_flat(64x65536) @ fc_w^T(65536x256), split-K=64
// with fp32 global atomics. acc pre-initialized with fc_b. Bandwidth-bound
// on the 64 MB fc_w stream.
// ---------------------------------------------------------------------------
__global__ __launch_bounds__(256) void fcinit_kernel(
    const float* __restrict__ fb, float* __restrict__ acc)
{
  int i = blockIdx.x * 256 + threadIdx.x;  // 16384
  acc[i] = fb[i & 255];
}

__global__ __launch_bounds__(256) void fc_kernel(
    const float* __restrict__ S, const float* __restrict__ fw,
    float* __restrict__ acc)
{
  const int tid = threadIdx.x;
  const int lane = tid & 31, wv = tid >> 5;
  const int nb = blockIdx.x & 3;      // N-tile (64 cols each)
  const int ksplit = blockIdx.x >> 2; // 0..63, 1024 K each
  const int n0 = nb * 64;
  const int kbase = ksplit * 1024;

  __shared__ float As[64 * 17];
  __shared__ float Bs[16 * 65];

  const int t0 = wv, t1 = wv + 8;
  const int tm0 = t0 >> 2, tn0 = t0 & 3, tm1 = t1 >> 2, tn1 = t1 & 3;
  const int lr = lane & 15;
  const int ah = (lane >> 4) * 2, bh = (lane >> 4);
  v8f acc0 = {}; v8f acc1 = {};

  for (int kk = kbase; kk < kbase + 1024; kk += 16) {
    __syncthreads();
    {
      int r = tid >> 2, kq = (tid & 3) * 4;
      // A: S_flat rows are batches (M=64)
      float4 va = *(const float4*)(S + (size_t)r * 65536 + kk + kq);
      As[r * 17 + kq + 0] = va.x; As[r * 17 + kq + 1] = va.y;
      As[r * 17 + kq + 2] = va.z; As[r * 17 + kq + 3] = va.w;
      // B = fc_w^T: Bs[k][n] = fc_w[n0+n][k]
      float4 vb = *(const float4*)(fw + (size_t)(n0 + r) * 65536 + kk + kq);
      Bs[(kq + 0) * 65 + r] = vb.x; Bs[(kq + 1) * 65 + r] = vb.y;
      Bs[(kq + 2) * 65 + r] = vb.z; Bs[(kq + 3) * 65 + r] = vb.w;
    }
    __syncthreads();
#pragma unroll
    for (int ks = 0; ks < 16; ks += 4) {
      v2f a0, a1, b0, b1;
      a0.x = As[(tm0 * 16 + lr) * 17 + ks + ah];
      a0.y = As[(tm0 * 16 + lr) * 17 + ks + ah + 1];
      b0.x = Bs[(ks + bh) * 65 + tn0 * 16 + lr];
      b0.y = Bs[(ks + bh + 2) * 65 + tn0 * 16 + lr];
      acc0 = wmma4(a0, b0, acc0);
      a1.x = As[(tm1 * 16 + lr) * 17 + ks + ah];
      a1.y = As[(tm1 * 16 + lr) * 17 + ks + ah + 1];
      b1.x = Bs[(ks + bh) * 65 + tn1 * 16 + lr];
      b1.y = Bs[(ks + bh + 2) * 65 + tn1 * 16 + lr];
      acc1 = wmma4(a1, b1, acc1);
    }
  }
  const int rhalf = (lane >> 4) * 8;
  V8 u0; u0.v = acc0; V8 u1; u1.v = acc1;
#pragma unroll
  for (int r = 0; r < 8; ++r) {
    int gb = tm0 * 16 + r + rhalf, gc = n0 + tn0 * 16 + lr;
    atomicAdd(&acc[gb * 256 + gc], u0.f[r]);
    gb = tm1 * 16 + r + rhalf; gc = n0 + tn1 * 16 + lr;
    atomicAdd(&acc[gb * 256 + gc], u1.f[r]);
  }
}

__global__ __launch_bounds__(256) void sig_kernel(
    const float* __restrict__ acc, float* __restrict__ att)
{
  int i = blockIdx.x * 256 + threadIdx.x;  // 16384
  att[i] = 1.0f / (1.0f + expf(-acc[i]));
}

// out[b][c][p] = x[b][c][p] * att[b][p]; float4 per thread.
__global__ __launch_bounds__(256) void scale_out_kernel(
    const float* __restrict__ x, const float* __restrict__ att,
    float* __restrict__ out)
{
  int idx = blockIdx.x * 256 + threadIdx.x;  // 4,194,304 float4s
  int base = idx * 4;
  int b = base >> 18;          // NC*NR = 262144 per batch
  int p = base & (NR - 1);     // rows are NR long, float4 never crosses
  const float* ab = att + b * NR + p;
  float4 xv = *(const float4*)(x + (size_t)base);
  float4 o;
  o.x = xv.x * ab[0]; o.y = xv.y * ab[1]; o.z = xv.z * ab[2]; o.w = xv.w * ab[3];
  *(float4*)(out + (size_t)base) = o;
}

// ---------------------------------------------------------------------------
extern "C" void kernel_launch(void* const* d_in, const int* in_sizes, int n_in,
                              void* d_out, int out_size, void* d_ws, size_t ws_size,
                              hipStream_t stream)
{
  (void)in_sizes; (void)n_in; (void)out_size; (void)ws_size;
  const float* x    = (const float*)d_in[0];
  const float* w_dr = (const float*)d_in[1];
  const float* b_dr = (const float*)d_in[2];
  const float* g1   = (const float*)d_in[3];
  const float* be1  = (const float*)d_in[4];
  const float* m1   = (const float*)d_in[5];
  const float* v1   = (const float*)d_in[6];
  const float* g2   = (const float*)d_in[7];
  const float* be2  = (const float*)d_in[8];
  const float* m2   = (const float*)d_in[9];
  const float* v2   = (const float*)d_in[10];
  const float* rcw  = (const float*)d_in[11];
  const float* rcb  = (const float*)d_in[12];
  const float* fw   = (const float*)d_in[13];
  const float* fb   = (const float*)d_in[14];
  float* out = (float*)d_out;

  // workspace carving (floats). Total ~27.3M floats (~105 MB).
  float* ws    = (float*)d_ws;
  float* z     = ws;  ws += (size_t)NB * NR * ND;   // 2,097,152
  float* mu    = ws;  ws += NB * NR;                // 16,384
  float* inorm = ws;  ws += NB;
  float* snorm = ws;  ws += NB;
  float* sc1   = ws;  ws += ND;
  float* sh1   = ws;  ws += ND;
  float* rsc   = ws;  ws += NR;
  float* rsh   = ws;  ws += NR;
  float* acc   = ws;  ws += NB * NR;                // 16,384
  float* att   = ws;  ws += NB * NR;                // 16,384
  const size_t MSZ = (size_t)NB * NR * NR;          // 4,194,304 each
  float* Amat = ws;  ws += MSZ;
  float* Tm   = ws;  ws += MSZ;
  float* Ya   = ws;  ws += MSZ;
  float* Yb   = ws;  ws += MSZ;
  float* Za   = ws;  ws += MSZ;
  float* Zb   = ws;  ws += MSZ;

  prep_kernel<<<1, 256, 0, stream>>>(b_dr, g1, be1, m1, v1, g2, be2, m2, v2,
                                     rcw, rcb, sc1, sh1, rsc, rsh);
  gemm1_kernel<<<NB * 8, 256, 0, stream>>>(x, w_dr, sc1, sh1, z);
  mean_kernel<<<NB * NR / 8, 256, 0, stream>>>(z, mu);
  cov_kernel<<<NB * 16, 256, 0, stream>>>(z, mu, Amat);
  trace_kernel<<<NB, 256, 0, stream>>>(Amat, inorm, snorm);

  // Newton-Schulz: Z0 = 0.5(3I - A_n); Y0 = A_n @ Z0
  initz_kernel<<<4096, 256, 0, stream>>>(Amat, inorm, Za);
  bmm256_kernel<<<NB * 4, 256, 0, stream>>>(Amat, Za, Ya, inorm, 1.0f, 0.0f);

  float* Y = Ya; float* Z = Za; float* Y2 = Yb; float* Z2 = Zb;
  for (int it = 0; it < 3; ++it) {  // iterN - 2
    bmm256_kernel<<<NB * 4, 256, 0, stream>>>(Z, Y, Tm, nullptr, -0.5f, 1.5f);
    bmm256_kernel<<<NB * 4, 256, 0, stream>>>(Y, Tm, Y2, nullptr, 1.0f, 0.0f);
    bmm256_kernel<<<NB * 4, 256, 0, stream>>>(Tm, Z, Z2, nullptr, 1.0f, 0.0f);
    float* t;
    t = Y; Y = Y2; Y2 = t;
    t = Z; Z = Z2; Z2 = t;
  }
  // Final: T = 0.5(3I - Z@Y);  S = (Y@T)*sqrt(normA), fused BN2+rowconv -> Amat
  bmm256_kernel<<<NB * 4, 256, 0, stream>>>(Z, Y, Tm, nullptr, -0.5f, 1.5f);
  bmm_final_kernel<<<NB * 4, 256, 0, stream>>>(Y, Tm, Amat, snorm, rsc, rsh);

  // FC + sigmoid + broadcast multiply
  fcinit_kernel<<<64, 256, 0, stream>>>(fb, acc);
  fc_kernel<<<256, 256, 0, stream>>>(Amat, fw, acc);
  sig_kernel<<<64, 256, 0, stream>>>(acc, att);
  scale_out_kernel<<<16384, 256, 0, stream>>>(x, att, out);
}